// SimpleGNNforOOD_55559696941389
// MI455X (gfx1250) — compile-verified
//
#include <hip/hip_runtime.h>
#include <hip/hip_bf16.h>

typedef float v2f __attribute__((ext_vector_type(2)));
typedef float v8f __attribute__((ext_vector_type(8)));

#define HIDF 128
#define NCLS 10

// ---------------- utility kernels ----------------

__global__ void k_zero(float* __restrict__ p, long long n) {
  long long i = (long long)blockIdx.x * blockDim.x + threadIdx.x;
  if (i < n) p[i] = 0.0f;
}

__global__ void k_init_deg(float* __restrict__ deg, int n) {
  int i = blockIdx.x * blockDim.x + threadIdx.x;
  if (i < n) deg[i] = 1.0f;  // self-loop contributes 1 to every node's degree
}

__global__ void k_deg_accum(const int* __restrict__ dst, float* __restrict__ deg, int ne) {
  int e = blockIdx.x * blockDim.x + threadIdx.x;
  if (e < ne) atomicAdd(&deg[dst[e]], 1.0f);
}

__global__ void k_deg_to_dis(float* __restrict__ d, int n) {
  int i = blockIdx.x * blockDim.x + threadIdx.x;
  if (i < n) d[i] = rsqrtf(d[i]);  // deg >= 1 always (self-loops)
}

// ---------------- WMMA fp32 GEMM: C[N,128] = A[N,128] @ W[128,128] ----------------
// One wave32 per 16x16 output tile; 8 waves/block cover 16 rows x 128 cols.
// K accumulated with 32 x V_WMMA_F32_16X16X4_F32 (exact fp32 FMA path).

__global__ __launch_bounds__(256)
void k_gemm128_wmma(const float* __restrict__ A,
                    const float* __restrict__ W,
                    float* __restrict__ C, int nrows) {
  const int wave = threadIdx.x >> 5;
  const int lane = threadIdx.x & 31;
  const int half = lane >> 4;   // 0: lanes 0-15, 1: lanes 16-31
  const int lrow = lane & 15;
  const int row0 = blockIdx.x << 4;
  const int col0 = wave << 4;
  if (row0 >= nrows) return;

  int ar = row0 + lrow;
  if (ar >= nrows) ar = nrows - 1;   // clamped dummy row (writes are guarded)
  const float* __restrict__ arow = A + (size_t)ar * HIDF;
  const float* __restrict__ wcol = W + col0 + lrow;

  v8f acc = {};
#pragma unroll
  for (int k0 = 0; k0 < HIDF; k0 += 4) {
    const int ka = k0 + 2 * half;
    v2f a, b;
    // A 16x4 f32: lanes 0-15 -> K = k0,k0+1 ; lanes 16-31 -> K = k0+2,k0+3
    a.x = arow[ka];
    a.y = arow[ka + 1];
    // B 4x16 f32: same K split across lane halves, N striped across lanes
    b.x = wcol[(size_t)ka * HIDF];
    b.y = wcol[(size_t)(ka + 1) * HIDF];
    acc = __builtin_amdgcn_wmma_f32_16x16x4_f32(false, a, false, b,
                                                (short)0, acc, false, false);
  }
#pragma unroll
  for (int v = 0; v < 8; ++v) {
    int r = row0 + v + 8 * half;   // C/D layout: VGPR v -> row v (+8 for upper half)
    if (r < nrows) C[(size_t)r * HIDF + col0 + lrow] = acc[v];
  }
}

// ---------------- edge scatter: agg[dst] += h[src] * dis[src]*dis[dst] ----------------
// One wave32 per edge; float4 gather, 4 L2-resident f32 atomics per lane.

__global__ __launch_bounds__(256)
void k_scatter(const float* __restrict__ h,
               const int* __restrict__ src, const int* __restrict__ dst,
               const float* __restrict__ dis,
               float* __restrict__ agg, int ne) {
  int e = (int)(((long long)blockIdx.x * blockDim.x + threadIdx.x) >> 5);
  int lane = threadIdx.x & 31;
  if (e >= ne) return;
  int s = src[e], d = dst[e];
  float nrm = dis[s] * dis[d];
  float4 m = ((const float4*)(h + (size_t)s * HIDF))[lane];
  float* ap = agg + (size_t)d * HIDF + lane * 4;
  atomicAdd(ap + 0, m.x * nrm);
  atomicAdd(ap + 1, m.y * nrm);
  atomicAdd(ap + 2, m.z * nrm);
  atomicAdd(ap + 3, m.w * nrm);
}

// ---------------- epilogue: agg = relu(agg + h*dis^2 + bias) ----------------
// Folds the self-loop term (norm = dis[i]^2) so no atomics are needed for it.

__global__ void k_bias_relu_self(float* __restrict__ agg, const float* __restrict__ h,
                                 const float* __restrict__ dis,
                                 const float* __restrict__ bias, int n) {
  long long idx = (long long)blockIdx.x * blockDim.x + threadIdx.x;
  if (idx >= (long long)n * HIDF) return;
  int i = (int)(idx >> 7);
  int f = (int)(idx & (HIDF - 1));
  float di = dis[i];
  float v = agg[idx] + h[idx] * di * di + bias[f];
  agg[idx] = v > 0.0f ? v : 0.0f;
}

// ---------------- pooling ----------------

__global__ __launch_bounds__(256)
void k_pool(const float* __restrict__ h, const int* __restrict__ batch,
            float* __restrict__ psum, float* __restrict__ pcnt, int n) {
  int i = (int)(((long long)blockIdx.x * blockDim.x + threadIdx.x) >> 5);
  int lane = threadIdx.x & 31;
  if (i >= n) return;
  int g = batch[i];
  float4 m = ((const float4*)(h + (size_t)i * HIDF))[lane];
  float* pp = psum + (size_t)g * HIDF + lane * 4;
  atomicAdd(pp + 0, m.x);
  atomicAdd(pp + 1, m.y);
  atomicAdd(pp + 2, m.z);
  atomicAdd(pp + 3, m.w);
  if (lane == 0) atomicAdd(&pcnt[g], 1.0f);
}

// ---------------- finalize: mean pool + classifier head ----------------

__global__ __launch_bounds__(HIDF)
void k_finalize(const float* __restrict__ psum, const float* __restrict__ pcnt,
                const float* __restrict__ Wc, const float* __restrict__ bc,
                float* __restrict__ out_logits, float* __restrict__ out_pooled) {
  __shared__ float p[HIDF];
  int g = blockIdx.x;
  int t = threadIdx.x;
  float c = pcnt[g];
  c = c > 1.0f ? c : 1.0f;
  float v = psum[(size_t)g * HIDF + t] / c;
  p[t] = v;
  out_pooled[(size_t)g * HIDF + t] = v;
  __syncthreads();
  if (t < NCLS) {
    float acc = bc[t];
#pragma unroll 8
    for (int k = 0; k < HIDF; ++k) acc += p[k] * Wc[k * NCLS + t];
    out_logits[g * NCLS + t] = acc;
  }
}

// ---------------- driver ----------------

extern "C" void kernel_launch(void* const* d_in, const int* in_sizes, int n_in,
                              void* d_out, int out_size, void* d_ws, size_t ws_size,
                              hipStream_t stream) {
  const float* x  = (const float*)d_in[0];
  const int*   ei = (const int*)d_in[1];
  const int*   batch = (const int*)d_in[2];
  const float* W1 = (const float*)d_in[3];
  const float* b1 = (const float*)d_in[4];
  const float* W2 = (const float*)d_in[5];
  const float* b2 = (const float*)d_in[6];
  const float* Wc = (const float*)d_in[7];
  const float* bc = (const float*)d_in[8];

  const int N = in_sizes[2];           // batch has N elements
  const int E = in_sizes[1] / 2;       // edge_index is [2,E]
  const int G = out_size / (NCLS + HIDF);

  const int* src = ei;
  const int* dst = ei + E;

  float* dis  = (float*)d_ws;                      // N
  float* bufA = dis + N;                           // N*128
  float* bufB = bufA + (size_t)N * HIDF;           // N*128
  float* psum = bufB + (size_t)N * HIDF;           // G*128
  float* pcnt = psum + (size_t)G * HIDF;           // G (contiguous after psum)

  float* out_logits = (float*)d_out;
  float* out_pooled = out_logits + (size_t)G * NCLS;

  const int T = 256;
  const long long nh = (long long)N * HIDF;
  const int gN  = (N + T - 1) / T;
  const int gE  = (E + T - 1) / T;
  const int gNH = (int)((nh + T - 1) / T);
  const int gEw = (int)(((long long)E * 32 + T - 1) / T);
  const int gNw = (int)(((long long)N * 32 + T - 1) / T);
  const int gMM = (N + 15) / 16;

  // degree -> dis = deg^{-1/2}
  k_init_deg<<<gN, T, 0, stream>>>(dis, N);
  k_deg_accum<<<gE, T, 0, stream>>>(dst, dis, E);
  k_deg_to_dis<<<gN, T, 0, stream>>>(dis, N);

  // layer 1: bufA = x@W1 ; bufB = relu(scatter(bufA) + self + b1)
  k_gemm128_wmma<<<gMM, T, 0, stream>>>(x, W1, bufA, N);
  k_zero<<<gNH, T, 0, stream>>>(bufB, nh);
  k_scatter<<<gEw, T, 0, stream>>>(bufA, src, dst, dis, bufB, E);
  k_bias_relu_self<<<gNH, T, 0, stream>>>(bufB, bufA, dis, b1, N);

  // layer 2: bufA = h1@W2 ; bufB = relu(scatter(bufA) + self + b2)
  k_gemm128_wmma<<<gMM, T, 0, stream>>>(bufB, W2, bufA, N);
  k_zero<<<gNH, T, 0, stream>>>(bufB, nh);
  k_scatter<<<gEw, T, 0, stream>>>(bufA, src, dst, dis, bufB, E);
  k_bias_relu_self<<<gNH, T, 0, stream>>>(bufB, bufA, dis, b2, N);

  // mean pool + classifier
  k_zero<<<(int)(((long long)G * HIDF + G + T - 1) / T), T, 0, stream>>>(psum, (long long)G * HIDF + G);
  k_pool<<<gNw, T, 0, stream>>>(bufB, batch, psum, pcnt, N);
  k_finalize<<<G, HIDF, 0, stream>>>(psum, pcnt, Wc, bc, out_logits, out_pooled);
}